// DiffusionMLS_88510686036697
// MI455X (gfx1250) — compile-verified
//
#include <hip/hip_runtime.h>
#include <stdint.h>

// Graph-Laplacian scatter: out[row] += w * (u[col] - u[row]) over 1.6M edges,
// F=32 features. Memory/atomic-bound (u/out are L2-resident on MI455X's 192MB
// L2). Edge streams staged via CDNA5 async global->LDS DMA (ASYNCcnt path);
// wave-uniform edge values promoted to SGPRs for saddr-form gathers/atomics.

#define TILE    512     // edges per block; 1,600,000 % 512 == 0
#define THREADS 256     // 8 wave32 waves per block
#define NFEAT   32      // lane == feature

__device__ __forceinline__ uint32_t lds_lo32(const void* p) {
  // Flat pointers to LDS carry the wave-relative LDS byte offset in the low
  // 32 bits (ISA 10.2); async-LDS VDST wants that.
  return (uint32_t)(uintptr_t)p;
}

__global__ void DiffusionMLS_zero_kernel(float4* __restrict__ p4,
                                         float* __restrict__ p, int n) {
  // Vectorized b128 zero-fill with scalar tail.
  int i = blockIdx.x * blockDim.x + threadIdx.x;
  int n4 = n >> 2;
  if (i < n4) p4[i] = make_float4(0.f, 0.f, 0.f, 0.f);
  int tail = n & 3;
  if (i < tail) p[(n4 << 2) + i] = 0.f;
}

__global__ __launch_bounds__(THREADS)
void DiffusionMLS_scatter_kernel(const float* __restrict__ u,
                                 const float* __restrict__ w,
                                 const int*   __restrict__ row,
                                 const int*   __restrict__ col,
                                 float*       __restrict__ out,
                                 int n_edges) {
  __shared__ alignas(16) int   s_row[TILE];
  __shared__ alignas(16) int   s_col[TILE];
  __shared__ alignas(16) float s_w[TILE];

  const int tid       = threadIdx.x;
  const int tile_base = blockIdx.x * TILE;
  const int rem       = min(TILE, n_edges - tile_base);

  if (rem == TILE) {
    // Full tile: async DMA global->LDS. Each of the 256 threads moves 8 bytes
    // of each array (2 int32 / 2 float) => 2048B per array == TILE elements.
    uint32_t l_row = lds_lo32(s_row) + (uint32_t)tid * 8u;
    uint32_t l_col = lds_lo32(s_col) + (uint32_t)tid * 8u;
    uint32_t l_w   = lds_lo32(s_w)   + (uint32_t)tid * 8u;
    uint32_t voff  = (uint32_t)tile_base * 4u + (uint32_t)tid * 8u;
    uint64_t b_row = (uint64_t)(uintptr_t)row;
    uint64_t b_col = (uint64_t)(uintptr_t)col;
    uint64_t b_w   = (uint64_t)(uintptr_t)w;

    asm volatile("global_load_async_to_lds_b64 %0, %1, %2 offset:0"
                 :: "v"(l_row), "v"(voff), "s"(b_row) : "memory");
    asm volatile("global_load_async_to_lds_b64 %0, %1, %2 offset:0"
                 :: "v"(l_col), "v"(voff), "s"(b_col) : "memory");
    asm volatile("global_load_async_to_lds_b64 %0, %1, %2 offset:0"
                 :: "v"(l_w),   "v"(voff), "s"(b_w)   : "memory");
    // Wait for this wave's async-LDS transfers (ASYNCcnt) to land.
    asm volatile("s_wait_asynccnt 0x0" ::: "memory");
  } else {
    // Generic guarded tail path (unused for E=1.6M, kept for safety).
    for (int i = tid; i < rem; i += THREADS) {
      s_row[i] = row[tile_base + i];
      s_col[i] = col[tile_base + i];
      s_w[i]   = w[tile_base + i];
    }
  }
  __syncthreads();

  // lane == feature: per edge, 32 lanes issue one coalesced 128B gather per
  // node row and one coalesced 128B burst of global_atomic_add_f32. Edge
  // values are wave-uniform -> readfirstlane to SGPRs so the gathers and
  // atomics use the scalar-base + lane-offset (saddr) addressing form.
  const int wave = tid >> 5;
  const int lane = tid & 31;
#pragma unroll 2
  for (int e = wave; e < rem; e += (THREADS / 32)) {
    const int rs = __builtin_amdgcn_readfirstlane(s_row[e]);
    const int cs = __builtin_amdgcn_readfirstlane(s_col[e]);
    const float wt = s_w[e];
    const float* __restrict__ uc = u + (size_t)cs * NFEAT;
    const float* __restrict__ ur = u + (size_t)rs * NFEAT;
    float* __restrict__ orow = out + (size_t)rs * NFEAT;
    const float d = uc[lane] - ur[lane];
    atomicAdd(&orow[lane], wt * d);
  }
}

extern "C" void kernel_launch(void* const* d_in, const int* in_sizes, int n_in,
                              void* d_out, int out_size, void* d_ws, size_t ws_size,
                              hipStream_t stream) {
  (void)n_in; (void)d_ws; (void)ws_size;
  const float* u  = (const float*)d_in[0];   // [N, 32]
  const float* w  = (const float*)d_in[1];   // [E]
  const int*   ei = (const int*)d_in[2];     // [2, E]: rows then cols
  const int n_edges = in_sizes[1];
  const int* row = ei;
  const int* col = ei + n_edges;
  float* out = (float*)d_out;                // [N, 32]

  // Harness poisons d_out; atomics accumulate, so zero first (same stream).
  const int zthreads = (out_size / 4) + 4;   // float4 lanes + tail cover
  DiffusionMLS_zero_kernel<<<(zthreads + 255) / 256, 256, 0, stream>>>(
      (float4*)out, out, out_size);

  const int blocks = (n_edges + TILE - 1) / TILE;
  DiffusionMLS_scatter_kernel<<<blocks, THREADS, 0, stream>>>(u, w, row, col, out, n_edges);
}